// SupConLossPrototype_22514218566252
// MI455X (gfx1250) — compile-verified
//
#include <hip/hip_runtime.h>

typedef float v2f __attribute__((ext_vector_type(2)));
typedef float v8f __attribute__((ext_vector_type(8)));

// Problem shape from the reference (B read at launch time).
constexpr int Tn = 1024;
constexpr int Hn = 256;
constexpr int Ln = 16;
constexpr float TEMP      = 0.07f;
constexpr float BASE_TEMP = 0.07f;
constexpr float NEG_INF_V = -1e30f;

__global__ void supcon_zero_out(float* out) { out[0] = 0.0f; }

__global__ __launch_bounds__(256)
void supcon_proto_kernel(const float* __restrict__ features,
                         const int*   __restrict__ lengths,
                         const int*   __restrict__ labels,
                         float*       __restrict__ out,
                         int B) {
  // LDS: accumulate layout [L][H] (conflict-free atomics),
  //      transposed layout [H][L] (conflict-free WMMA B-operand loads),
  //      per-wave logits staging padded to 17 (17 coprime 64 -> no bank conflicts).
  __shared__ float proto[Ln][Hn];      // 16 KB
  __shared__ float protoT[Hn][Ln];     // 16 KB
  __shared__ float counts[Ln];
  __shared__ float ltile[8][16][17];   // ~8.5 KB
  __shared__ float block_loss;

  const int b    = blockIdx.x;
  const int tid  = threadIdx.x;
  const int wave = tid >> 5;           // 0..7 (wave32)
  const int lane = tid & 31;
  const int len  = lengths[b];
  const float* fb = features + (size_t)b * Tn * Hn;
  const int*   lb = labels   + (size_t)b * Tn;

  // ---- zero LDS accumulators ----
  for (int i = tid; i < Ln * Hn; i += 256) (&proto[0][0])[i] = 0.0f;
  if (tid < Ln) counts[tid] = 0.0f;
  if (tid == 0) block_loss = 0.0f;
  __syncthreads();

  // ---- Phase 1: proto_sum[L][H] and counts via LDS float atomics ----
  // Wave w handles tokens w, w+8, ...; lane covers h = lane + 32k  -> 32 distinct banks.
  for (int t = wave; t < len; t += 8) {
    const int l = lb[t];
    const float* src = fb + (size_t)t * Hn;
    float* dst = &proto[l][0];
#pragma unroll
    for (int k = 0; k < Hn / 32; ++k) {
      atomicAdd(&dst[lane + 32 * k], src[lane + 32 * k]);   // ds_add_f32
    }
    if (lane == 0) atomicAdd(&counts[l], 1.0f);
  }
  __syncthreads();

  // ---- prototypes = proto_sum / max(count,1), stored transposed [H][L] ----
  for (int i = tid; i < Ln * Hn; i += 256) {
    const int l = i >> 8;        // / Hn
    const int h = i & (Hn - 1);
    const float c = counts[l];
    protoT[h][l] = proto[l][h] * (1.0f / fmaxf(c, 1.0f));
  }
  __syncthreads();

  // ---- Phase 2: logits via V_WMMA_F32_16X16X4_F32, then masked log-softmax ----
  // A (16x4 f32) layout: lanes 0-15 -> M=lane, K=0,1 ; lanes 16-31 -> M=lane-16, K=2,3
  // B (4x16 f32) layout: vgpr j, lanes 0-15 -> K=j, N=lane ; lanes 16-31 -> K=2+j, N=lane-16
  const int  half = lane >> 4;       // 0 or 1
  const int  mrow = lane & 15;       // token row within tile / label column
  const int  koff = half * 2;        // K offset for this half-wave
  const float inv_temp = 1.0f / TEMP;
  const float ratio    = TEMP / BASE_TEMP;
  float loss_acc = 0.0f;

  for (int t0 = wave * 16; t0 < Tn; t0 += 8 * 16) {   // 8 tiles per wave, uniform trip count
    v8f acc = {0.f, 0.f, 0.f, 0.f, 0.f, 0.f, 0.f, 0.f};
    const float* arow = fb + (size_t)(t0 + mrow) * Hn + koff;   // 8B-aligned
#pragma unroll 8
    for (int h0 = 0; h0 < Hn; h0 += 4) {
      v2f a  = *(const v2f*)(arow + h0);                 // global_load_b64 (L2-hot)
      v2f bb = { protoT[h0 + koff][mrow],                // conflict-free ds_load_b32 x2
                 protoT[h0 + koff + 1][mrow] };
      acc = __builtin_amdgcn_wmma_f32_16x16x4_f32(
          /*neg_a=*/false, a, /*neg_b=*/false, bb,
          /*c_mod=*/(short)0, acc, /*reuse_a=*/false, /*reuse_b=*/false);
    }
    // C layout: vgpr i, lanes 0-15 -> M=i, N=lane ; lanes 16-31 -> M=8+i, N=lane-16
#pragma unroll
    for (int i = 0; i < 8; ++i)
      ltile[wave][i + half * 8][mrow] = acc[i];
    __syncthreads();

    if (lane < 16) {
      const int t = t0 + lane;
      if (t < len) {
        const int l = lb[t];
        float row[Ln];
        float mx = NEG_INF_V;
#pragma unroll
        for (int n = 0; n < Ln; ++n) {
          float v = ltile[wave][lane][n] * inv_temp;
          v = (counts[n] > 0.0f) ? v : NEG_INF_V;        // mask absent labels
          row[n] = v;
          mx = fmaxf(mx, v);
        }
        float s = 0.0f;
#pragma unroll
        for (int n = 0; n < Ln; ++n) s += __expf(row[n] - mx);
        const float lse = mx + __logf(s);
        loss_acc += -ratio * (row[l] - lse);             // -log_prob at own label
      }
    }
    __syncthreads();
  }

  atomicAdd(&block_loss, loss_acc);
  __syncthreads();
  if (tid == 0) {
    const float per_dialogue = block_loss / (float)len;
    atomicAdd(out, per_dialogue / (float)B);
  }
}

extern "C" void kernel_launch(void* const* d_in, const int* in_sizes, int n_in,
                              void* d_out, int out_size, void* d_ws, size_t ws_size,
                              hipStream_t stream) {
  const float* features = (const float*)d_in[0];
  const int*   lengths  = (const int*)d_in[1];
  const int*   labels   = (const int*)d_in[2];
  float*       out      = (float*)d_out;
  const int B = in_sizes[1];   // number of dialogues

  supcon_zero_out<<<1, 1, 0, stream>>>(out);
  supcon_proto_kernel<<<B, 256, 0, stream>>>(features, lengths, labels, out, B);
}